// StructureAwareAttention_50199577756224
// MI455X (gfx1250) — compile-verified
//
#include <hip/hip_runtime.h>
#include <math.h>

#define N_NODES   50000
#define C_DIM     128
#define E_EDGES   800000
#define HEADS     4
#define HEAD_DIM  32
#define LN_EPS    1e-5f

typedef __attribute__((ext_vector_type(2))) float v2f;
typedef __attribute__((ext_vector_type(8))) float v8f;

// ---------------------------------------------------------------------------
// CAS-based float atomic max (handles negatives correctly)
// ---------------------------------------------------------------------------
__device__ __forceinline__ void atomicMaxFloat(float* addr, float val) {
    int* ai = (int*)addr;
    int old = __float_as_int(*addr);
    while (__int_as_float(old) < val) {
        int assumed = old;
        old = atomicCAS(ai, assumed, __float_as_int(val));
        if (old == assumed) break;
    }
}

// ---------------------------------------------------------------------------
// Async global->LDS 16B copy (gfx1250 GLOBAL_LOAD_ASYNC_TO_LDS_B128, ASYNCcnt)
// lds_off: byte offset into workgroup LDS (low 32 bits of flat shared addr)
// ---------------------------------------------------------------------------
__device__ __forceinline__ void async_copy_b128(unsigned lds_off, const void* gptr) {
    asm volatile("global_load_async_to_lds_b128 %0, %1, off"
                 :: "v"(lds_off), "v"(gptr)
                 : "memory");
}
__device__ __forceinline__ void async_wait0() {
    asm volatile("s_wait_asynccnt 0x0" ::: "memory");
}

// ---------------------------------------------------------------------------
// Kernel 0: init m = -inf, s = 0, acc = 0 (vectorized zero-fill)
// ---------------------------------------------------------------------------
__global__ __launch_bounds__(256) void init_kernel(float* __restrict__ mmax,
                                                   float* __restrict__ ssum,
                                                   float* __restrict__ acc) {
    int tid = blockIdx.x * 256 + threadIdx.x;
    if (tid < (N_NODES * C_DIM) / 4) {
        float4 z = make_float4(0.f, 0.f, 0.f, 0.f);
        ((float4*)acc)[tid] = z;
    }
    if (tid < N_NODES * HEADS) { mmax[tid] = -3.0e38f; ssum[tid] = 0.0f; }
}

// ---------------------------------------------------------------------------
// Kernel 1: Q/K/V = x @ W{q,k,v} + b   (fp32 WMMA 16x16x4, async A-staging)
// Block = 16 rows of x; 8 waves x 3 column-tiles = 24 tiles = 3 matrices x 8.
// ---------------------------------------------------------------------------
__global__ __launch_bounds__(256) void qkv_gemm_kernel(
    const float* __restrict__ x,
    const float* __restrict__ Wq, const float* __restrict__ bq,
    const float* __restrict__ Wk, const float* __restrict__ bk,
    const float* __restrict__ Wv, const float* __restrict__ bv,
    float* __restrict__ q, float* __restrict__ k, float* __restrict__ v)
{
    __shared__ float xs[16][C_DIM + 4];
    const int rowbase = blockIdx.x * 16;
    const int tid = threadIdx.x;

    // async-stage 16x128 fp32 tile as 512 x b128 chunks (row r, 16B chunk c)
    #pragma unroll
    for (int it = 0; it < 2; ++it) {
        int idx = tid + it * 256;
        int r = idx >> 5, c = idx & 31;
        async_copy_b128((unsigned)(size_t)&xs[r][c * 4],
                        x + (size_t)(rowbase + r) * C_DIM + c * 4);
    }
    async_wait0();
    __syncthreads();

    const int wave = tid >> 5;
    const int lane = tid & 31;
    const int hs   = lane >> 4;   // half-wave select (K pair / row+8)
    const int li   = lane & 15;

    for (int j = 0; j < 3; ++j) {
        const int t   = wave * 3 + j;       // 0..23
        const int mat = t >> 3;             // 0=q 1=k 2=v
        const int ncb = (t & 7) * 16;       // column tile base
        const float* W    = (mat == 0) ? Wq : (mat == 1) ? Wk : Wv;
        const float* bias = (mat == 0) ? bq : (mat == 1) ? bk : bv;
        float*       out  = (mat == 0) ? q  : (mat == 1) ? k  : v;

        v8f c8 = {};
        for (int kb = 0; kb < C_DIM; kb += 4) {
            const int k0 = kb + hs * 2;
            v2f a; a.x = xs[li][k0];                 a.y = xs[li][k0 + 1];
            v2f b; b.x = W[k0 * C_DIM + ncb + li];   b.y = W[(k0 + 1) * C_DIM + ncb + li];
            c8 = __builtin_amdgcn_wmma_f32_16x16x4_f32(false, a, false, b,
                                                       (short)0, c8, false, false);
        }
        const float bval = bias[ncb + li];
        #pragma unroll
        for (int p = 0; p < 8; ++p) {
            const int row = rowbase + p + hs * 8;
            out[row * C_DIM + ncb + li] = c8[p] + bval;
        }
    }
}

// ---------------------------------------------------------------------------
// Kernel 2: per (edge, head) attention score + segment max
// ---------------------------------------------------------------------------
__global__ __launch_bounds__(256) void edge_score_kernel(
    const float* __restrict__ q, const float* __restrict__ k,
    const int* __restrict__ ei, const int* __restrict__ stype,
    const float* __restrict__ sbias,
    float* __restrict__ attn, float* __restrict__ mmax)
{
    const int tid = blockIdx.x * 256 + threadIdx.x;
    if (tid >= E_EDGES * HEADS) return;
    const int e = tid >> 2;
    const int h = tid & 3;
    const int src = ei[e];
    const int dst = ei[E_EDGES + e];

    const float4* qp = (const float4*)(q + (size_t)dst * C_DIM + h * HEAD_DIM);
    const float4* kp = (const float4*)(k + (size_t)src * C_DIM + h * HEAD_DIM);
    float acc = 0.0f;
    #pragma unroll
    for (int i = 0; i < HEAD_DIM / 4; ++i) {
        float4 qv = qp[i], kv = kp[i];
        acc += qv.x * kv.x + qv.y * kv.y + qv.z * kv.z + qv.w * kv.w;
    }
    float sc = acc * 0.17677669529663689f;          // 1/sqrt(32)
    sc += sbias[stype[src] * HEADS + h];
    attn[tid] = sc;
    atomicMaxFloat(&mmax[dst * HEADS + h], sc);
}

// ---------------------------------------------------------------------------
// Kernel 3: e = exp(score - m[dst]); s[dst] += e; acc[dst,:] += e * v[src,:]
// ---------------------------------------------------------------------------
__global__ __launch_bounds__(256) void edge_scatter_kernel(
    const float* __restrict__ v, const int* __restrict__ ei,
    const float* __restrict__ attn, const float* __restrict__ mmax,
    float* __restrict__ ssum, float* __restrict__ acc)
{
    const int tid = blockIdx.x * 256 + threadIdx.x;
    if (tid >= E_EDGES * HEADS) return;
    const int e = tid >> 2;
    const int h = tid & 3;
    const int src = ei[e];
    const int dst = ei[E_EDGES + e];

    const float ev = __expf(attn[tid] - mmax[dst * HEADS + h]);
    atomicAdd(&ssum[dst * HEADS + h], ev);

    const float4* vp = (const float4*)(v + (size_t)src * C_DIM + h * HEAD_DIM);
    float* ap = acc + (size_t)dst * C_DIM + h * HEAD_DIM;
    #pragma unroll
    for (int i = 0; i < HEAD_DIM / 4; ++i) {
        float4 vv = vp[i];
        atomicAdd(&ap[4 * i + 0], ev * vv.x);
        atomicAdd(&ap[4 * i + 1], ev * vv.y);
        atomicAdd(&ap[4 * i + 2], ev * vv.z);
        atomicAdd(&ap[4 * i + 3], ev * vv.w);
    }
}

// ---------------------------------------------------------------------------
// Kernel 4: out = LN( x + (acc/(s+eps)) @ Wo + bo )
// fp32 WMMA; acc tile async-copied raw, softmax normalization folded into the
// A-fragment read via per-(row,head) reciprocals; fused bias+residual+LN.
// ---------------------------------------------------------------------------
__global__ __launch_bounds__(256) void out_gemm_ln_kernel(
    const float* __restrict__ acc, const float* __restrict__ ssum,
    const float* __restrict__ Wo,  const float* __restrict__ bo,
    const float* __restrict__ x,
    const float* __restrict__ gamma, const float* __restrict__ beta,
    float* __restrict__ out)
{
    __shared__ float as[16][C_DIM + 4];
    __shared__ float ys[16][C_DIM + 4];
    __shared__ float sinv[16][HEADS];
    __shared__ float mu_s[16], rstd_s[16];

    const int rowbase = blockIdx.x * 16;
    const int tid = threadIdx.x;

    // async-stage raw acc tile (16x128 fp32) as 512 x b128 chunks
    #pragma unroll
    for (int it = 0; it < 2; ++it) {
        int idx = tid + it * 256;
        int r = idx >> 5, c = idx & 31;
        async_copy_b128((unsigned)(size_t)&as[r][c * 4],
                        acc + (size_t)(rowbase + r) * C_DIM + c * 4);
    }
    // per-(row, head) softmax reciprocal
    if (tid < 16 * HEADS) {
        int r = tid >> 2, h = tid & 3;
        sinv[r][h] = 1.0f / (ssum[(rowbase + r) * HEADS + h] + 1e-16f);
    }
    async_wait0();
    __syncthreads();

    const int wave = tid >> 5;
    const int lane = tid & 31;
    const int hs   = lane >> 4;
    const int li   = lane & 15;
    const int ncb  = wave * 16;

    v8f c8 = {};
    for (int kb = 0; kb < C_DIM; kb += 4) {
        const int k0 = kb + hs * 2;
        const float rs = sinv[li][k0 >> 5];          // k0,k0+1 in same head block
        v2f a; a.x = as[li][k0] * rs;             a.y = as[li][k0 + 1] * rs;
        v2f b; b.x = Wo[k0 * C_DIM + ncb + li];   b.y = Wo[(k0 + 1) * C_DIM + ncb + li];
        c8 = __builtin_amdgcn_wmma_f32_16x16x4_f32(false, a, false, b,
                                                   (short)0, c8, false, false);
    }
    const float bval = bo[ncb + li];
    #pragma unroll
    for (int p = 0; p < 8; ++p) {
        const int row = p + hs * 8;
        float y = c8[p] + bval + x[(size_t)(rowbase + row) * C_DIM + ncb + li];
        ys[row][ncb + li] = y;
    }
    __syncthreads();

    // LayerNorm stats: 8 lanes per row, shuffle tree reduction
    if (tid < 128) {
        const int r = tid >> 3, l8 = tid & 7;
        float sum = 0.0f, sq = 0.0f;
        #pragma unroll
        for (int j = 0; j < 16; ++j) {
            float yv = ys[r][l8 * 16 + j];
            sum += yv; sq += yv * yv;
        }
        #pragma unroll
        for (int off = 1; off < 8; off <<= 1) {
            sum += __shfl_xor(sum, off, 32);
            sq  += __shfl_xor(sq,  off, 32);
        }
        if (l8 == 0) {
            const float mu  = sum * (1.0f / C_DIM);
            const float var = sq * (1.0f / C_DIM) - mu * mu;
            mu_s[r]   = mu;
            rstd_s[r] = rsqrtf(var + LN_EPS);
        }
    }
    __syncthreads();

    // vectorized epilogue: (y - mu) * rstd * gamma + beta
    #pragma unroll
    for (int it = 0; it < 2; ++it) {
        int idx = tid + it * 256;
        int r = idx >> 5, c4 = (idx & 31) * 4;
        float4 yv = *(const float4*)&ys[r][c4];
        float4 gv = *(const float4*)&gamma[c4];
        float4 bv = *(const float4*)&beta[c4];
        const float mu = mu_s[r], rstd = rstd_s[r];
        float4 o;
        o.x = (yv.x - mu) * rstd * gv.x + bv.x;
        o.y = (yv.y - mu) * rstd * gv.y + bv.y;
        o.z = (yv.z - mu) * rstd * gv.z + bv.z;
        o.w = (yv.w - mu) * rstd * gv.w + bv.w;
        *(float4*)&out[(size_t)(rowbase + r) * C_DIM + c4] = o;
    }
}

// ---------------------------------------------------------------------------
// Launch
// ---------------------------------------------------------------------------
extern "C" void kernel_launch(void* const* d_in, const int* in_sizes, int n_in,
                              void* d_out, int out_size, void* d_ws, size_t ws_size,
                              hipStream_t stream) {
    const float* x     = (const float*)d_in[0];
    const int*   stype = (const int*)  d_in[1];
    const int*   ei    = (const int*)  d_in[2];
    const float* Wq    = (const float*)d_in[3];
    const float* bq    = (const float*)d_in[4];
    const float* Wk    = (const float*)d_in[5];
    const float* bk    = (const float*)d_in[6];
    const float* Wv    = (const float*)d_in[7];
    const float* bv    = (const float*)d_in[8];
    const float* sbias = (const float*)d_in[9];
    const float* Wo    = (const float*)d_in[10];
    const float* bo    = (const float*)d_in[11];
    const float* gam   = (const float*)d_in[12];
    const float* bet   = (const float*)d_in[13];
    float* out = (float*)d_out;

    // workspace layout (floats)
    const size_t NC = (size_t)N_NODES * C_DIM;     // 6.4M
    const size_t EH = (size_t)E_EDGES * HEADS;     // 3.2M
    const size_t NH = (size_t)N_NODES * HEADS;     // 200K
    float* ws   = (float*)d_ws;
    float* q    = ws;
    float* k    = q + NC;
    float* v    = k + NC;
    float* acc  = v + NC;
    float* attn = acc + NC;
    float* mmax = attn + EH;
    float* ssum = mmax + NH;

    const int nInit = (int)((NC / 4 + 255) / 256);
    init_kernel<<<nInit, 256, 0, stream>>>(mmax, ssum, acc);

    qkv_gemm_kernel<<<N_NODES / 16, 256, 0, stream>>>(x, Wq, bq, Wk, bk, Wv, bv, q, k, v);

    const int nEdge = (int)((EH + 255) / 256);
    edge_score_kernel<<<nEdge, 256, 0, stream>>>(q, k, ei, stype, sbias, attn, mmax);
    edge_scatter_kernel<<<nEdge, 256, 0, stream>>>(v, ei, attn, mmax, ssum, acc);

    out_gemm_ln_kernel<<<N_NODES / 16, 256, 0, stream>>>(acc, ssum, Wo, bo, x, gam, bet, out);
}